// Decoder_6700148982088
// MI455X (gfx1250) — compile-verified
//
#include <hip/hip_runtime.h>

typedef __attribute__((ext_vector_type(2))) float v2f;
typedef __attribute__((ext_vector_type(8))) float v8f;

#define BB 8
#define CC 512
#define CQ 64
#define HH 96
#define WW 96
#define HWN (HH * WW) // 9216

__device__ __forceinline__ v8f wmma4(v2f a, v2f b, v8f c) {
  // D = A(16x4,f32) * B(4x16,f32) + C(16x16,f32)
  return __builtin_amdgcn_wmma_f32_16x16x4_f32(
      /*neg_a=*/false, a, /*neg_b=*/false, b,
      /*c_mod=*/(short)0, c, /*reuse_a=*/false, /*reuse_b=*/false);
}

// --- CDNA5 async global->LDS copies (ASYNCcnt path) ------------------------
// LDS aperture: generic-pointer bits [31:0] are the LDS byte offset (ISA 10.2),
// which is exactly what the instruction's VDST operand wants.
__device__ __forceinline__ void async_ld_lds_b32(void* lds, const void* gptr) {
  asm volatile("global_load_async_to_lds_b32 %0, %1, off" ::"v"(
                   (unsigned)(uintptr_t)lds),
               "v"(gptr)
               : "memory");
}
__device__ __forceinline__ void async_ld_lds_b128(void* lds, const void* gptr) {
  asm volatile("global_load_async_to_lds_b128 %0, %1, off" ::"v"(
                   (unsigned)(uintptr_t)lds),
               "v"(gptr)
               : "memory");
}
__device__ __forceinline__ void wait_async() {
  asm volatile("s_wait_asynccnt 0x0" ::: "memory");
}

// ---------------------------------------------------------------------------
// Kernel 1: fused QKV projection.  Stacked output rows: [0,64)=q, [64,128)=k,
// [128,640)=v.  Per batch: Out[o][n] = sum_c W[o][c] * x[b][c][n] + bias[o].
// Wave computes a 16x64 tile (4 accumulators); block of 4 waves covers 64x64.
// ---------------------------------------------------------------------------
__global__ __launch_bounds__(128)
void qkv_proj_kernel(const float* __restrict__ x,
                     const float* __restrict__ Wq, const float* __restrict__ bq,
                     const float* __restrict__ Wk, const float* __restrict__ bk,
                     const float* __restrict__ Wv, const float* __restrict__ bv,
                     float* __restrict__ qo, float* __restrict__ ko,
                     float* __restrict__ vo) {
  const int lane = threadIdx.x & 31;
  const int wave = threadIdx.x >> 5;                 // 0..3
  const int b = blockIdx.z;
  const int nBase = blockIdx.x * 64;
  const int oBase = blockIdx.y * 64 + wave * 16;     // 16-row tile

  const float* Wsrc;
  const float* bsrc;
  float* osrc;
  int oLoc, chan;
  if (oBase < 64) {
    Wsrc = Wq; bsrc = bq; osrc = qo; oLoc = oBase;        chan = CQ;
  } else if (oBase < 128) {
    Wsrc = Wk; bsrc = bk; osrc = ko; oLoc = oBase - 64;   chan = CQ;
  } else {
    Wsrc = Wv; bsrc = bv; osrc = vo; oLoc = oBase - 128;  chan = CC;
  }

  const float* xb = x + (size_t)b * CC * HWN;
  const int mr = lane & 15;   // row-in-tile for A, col-in-tile for B
  const int hi = lane >> 4;   // K sub-pair select

  v8f zero = {};
  v8f acc0 = zero, acc1 = zero, acc2 = zero, acc3 = zero;

  for (int k0 = 0; k0 < CC; k0 += 4) {
    const int ka = k0 + 2 * hi;
    const float* wrow = Wsrc + (size_t)(oLoc + mr) * CC + ka;
    v2f a;
    a.x = wrow[0];
    a.y = wrow[1];
    const float* xk = xb + (size_t)ka * HWN + nBase + mr;
    // speculative prefetch of the x rows 4 K-iterations ahead (x is streamed
    // exactly once; pull it toward the WGP early).
    __builtin_prefetch(xk + 16 * HWN, 0, 3);
    v2f b0, b1, b2, b3;
    b0.x = xk[0];  b0.y = xk[HWN + 0];
    b1.x = xk[16]; b1.y = xk[HWN + 16];
    b2.x = xk[32]; b2.y = xk[HWN + 32];
    b3.x = xk[48]; b3.y = xk[HWN + 48];
    acc0 = wmma4(a, b0, acc0);
    acc1 = wmma4(a, b1, acc1);
    acc2 = wmma4(a, b2, acc2);
    acc3 = wmma4(a, b3, acc3);
  }

  float* outb = osrc + (size_t)b * chan * HWN;
#pragma unroll
  for (int i = 0; i < 8; ++i) {
    const int r = oLoc + i + 8 * hi;
    const float bias = bsrc[r];
    float* orow = outb + (size_t)r * HWN + nBase + mr;
    orow[0]  = acc0[i] + bias;
    orow[16] = acc1[i] + bias;
    orow[32] = acc2[i] + bias;
    orow[48] = acc3[i] + bias;
  }
}

// ---------------------------------------------------------------------------
// Kernel 2: column energies eH[b][h][w][p] = sum_c q[b,c,h,w]*k[b,c,p,w],
// diag (p==h) masked to -inf.  One workgroup per (w,b); 6 waves, each owns a
// 16-row h-tile over all 96 p columns (6 accumulators).  K-strip staged in
// LDS via async-to-LDS (strided gather -> b32 per element).
// ---------------------------------------------------------------------------
__global__ __launch_bounds__(192)
void energyH_kernel(const float* __restrict__ q, const float* __restrict__ k,
                    float* __restrict__ eH) {
  __shared__ float kst[CQ * HH];  // kst[c*96+p] = k[b,c,p,w]
  const int w = blockIdx.x, b = blockIdx.y;
  const int tid = threadIdx.x;
  const float* qb = q + (size_t)b * CQ * HWN;
  const float* kb = k + (size_t)b * CQ * HWN;

  for (int i = tid; i < CQ * HH; i += 192) {  // 32 full-EXEC iterations
    const int c = i / HH, p = i % HH;
    async_ld_lds_b32(&kst[i], kb + (size_t)c * HWN + p * WW + w);
  }
  wait_async();
  __syncthreads();

  const int lane = tid & 31;
  const int wave = tid >> 5;   // h-tile index 0..5
  const int mr = lane & 15, hi = lane >> 4;
  const int h0 = wave * 16;

  v8f zero = {};
  v8f acc[6];
#pragma unroll
  for (int j = 0; j < 6; ++j) acc[j] = zero;

  for (int k0 = 0; k0 < CQ; k0 += 4) {
    const int ka = k0 + 2 * hi;
    // A = Q^T : A[h][c] = q[b,c,h,w]
    const float* qp = qb + (size_t)ka * HWN + (h0 + mr) * WW + w;
    v2f a;
    a.x = qp[0];
    a.y = qp[HWN];
#pragma unroll
    for (int j = 0; j < 6; ++j) {
      const float* kp = &kst[ka * HH + j * 16 + mr];
      v2f bf;
      bf.x = kp[0];
      bf.y = kp[HH];
      acc[j] = wmma4(a, bf, acc[j]);
    }
  }

#pragma unroll
  for (int i = 0; i < 8; ++i) {
    const int h = h0 + i + 8 * hi;
#pragma unroll
    for (int j = 0; j < 6; ++j) {
      const int p = j * 16 + mr;
      const float val = (p == h) ? -__builtin_inff() : acc[j][i];
      eH[(((size_t)b * HH + h) * WW + w) * HH + p] = val;
    }
  }
}

// ---------------------------------------------------------------------------
// Kernel 3: row energies eW[b][h][w][p] = sum_c q[b,c,h,w]*k[b,c,h,p].
// K-strip is contiguous along p -> async b128 staging.
// ---------------------------------------------------------------------------
__global__ __launch_bounds__(192)
void energyW_kernel(const float* __restrict__ q, const float* __restrict__ k,
                    float* __restrict__ eW) {
  __shared__ float kst[CQ * WW];  // kst[c*96+p] = k[b,c,h,p]
  const int h = blockIdx.x, b = blockIdx.y;
  const int tid = threadIdx.x;
  const float* qb = q + (size_t)b * CQ * HWN;
  const float* kb = k + (size_t)b * CQ * HWN;

  for (int i = tid; i < (CQ * WW) / 4; i += 192) {  // 8 full-EXEC iterations
    const int e = i * 4;
    const int c = e / WW, p = e % WW;
    async_ld_lds_b128(&kst[e], kb + (size_t)c * HWN + h * WW + p);
  }
  wait_async();
  __syncthreads();

  const int lane = tid & 31;
  const int wave = tid >> 5;  // w-tile index 0..5
  const int mr = lane & 15, hi = lane >> 4;
  const int w0 = wave * 16;

  v8f zero = {};
  v8f acc[6];
#pragma unroll
  for (int j = 0; j < 6; ++j) acc[j] = zero;

  for (int k0 = 0; k0 < CQ; k0 += 4) {
    const int ka = k0 + 2 * hi;
    // A = Q^T : A[w][c] = q[b,c,h,w]
    const float* qp = qb + (size_t)ka * HWN + h * WW + (w0 + mr);
    v2f a;
    a.x = qp[0];
    a.y = qp[HWN];
#pragma unroll
    for (int j = 0; j < 6; ++j) {
      const float* kp = &kst[ka * WW + j * 16 + mr];
      v2f bf;
      bf.x = kp[0];
      bf.y = kp[WW];
      acc[j] = wmma4(a, bf, acc[j]);
    }
  }

#pragma unroll
  for (int i = 0; i < 8; ++i) {
    const int w = w0 + i + 8 * hi;
#pragma unroll
    for (int j = 0; j < 6; ++j) {
      const int p = j * 16 + mr;
      eW[(((size_t)b * HH + h) * WW + w) * WW + p] = acc[j][i];
    }
  }
}

// ---------------------------------------------------------------------------
// Kernel 4: softmax over the concatenated 192 logits per pixel, in place.
// One wave32 per pixel; 6 values per lane; shuffle reductions.
// ---------------------------------------------------------------------------
__global__ __launch_bounds__(256)
void softmax_kernel(float* __restrict__ eH, float* __restrict__ eW) {
  const int wave = threadIdx.x >> 5;
  const int lane = threadIdx.x & 31;
  const size_t pix = (size_t)blockIdx.x * 8 + wave;  // (b*96+h)*96+w
  float* rH = eH + pix * HH;
  float* rW = eW + pix * WW;

  float v0 = rH[lane], v1 = rH[lane + 32], v2 = rH[lane + 64];
  float v3 = rW[lane], v4 = rW[lane + 32], v5 = rW[lane + 64];

  float m = fmaxf(fmaxf(fmaxf(v0, v1), fmaxf(v2, v3)), fmaxf(v4, v5));
#pragma unroll
  for (int off = 16; off > 0; off >>= 1) m = fmaxf(m, __shfl_xor(m, off, 32));

  v0 = __expf(v0 - m); v1 = __expf(v1 - m); v2 = __expf(v2 - m);
  v3 = __expf(v3 - m); v4 = __expf(v4 - m); v5 = __expf(v5 - m);

  float s = ((v0 + v1) + (v2 + v3)) + (v4 + v5);
#pragma unroll
  for (int off = 16; off > 0; off >>= 1) s += __shfl_xor(s, off, 32);

  const float inv = 1.0f / s;
  rH[lane] = v0 * inv; rH[lane + 32] = v1 * inv; rH[lane + 64] = v2 * inv;
  rW[lane] = v3 * inv; rW[lane + 32] = v4 * inv; rW[lane + 64] = v5 * inv;
}

// ---------------------------------------------------------------------------
// Kernel 5: outH[b,c,h,w] = sum_p v[b,c,p,w] * attH[b,h,w,p]  (raw, no gamma).
// One workgroup per (w,b); attH strip (96x96, rows contiguous) staged in LDS
// via async b128; 8 waves sweep the 32 c-tiles, each 16x96 via 6 accumulators.
// ---------------------------------------------------------------------------
__global__ __launch_bounds__(256)
void aggH_kernel(const float* __restrict__ v, const float* __restrict__ attH,
                 float* __restrict__ out) {
  __shared__ float ast[HH * HH];  // ast[h*96+p]
  const int w = blockIdx.x, b = blockIdx.y;
  const int tid = threadIdx.x;
  const float* attb = attH + (((size_t)b * HH) * WW + w) * HH;

  for (int i = tid; i < (HH * HH) / 4; i += 256) {  // 9 full-EXEC iterations
    const int e = i * 4;
    const int h = e / HH, p = e % HH;
    async_ld_lds_b128(&ast[e], attb + (size_t)h * WW * HH + p);
  }
  wait_async();
  __syncthreads();

  const int lane = tid & 31, wave = tid >> 5;
  const int mr = lane & 15, hi = lane >> 4;
  const float* vb = v + (size_t)b * CC * HWN;
  float* ob = out + (size_t)b * CC * HWN;

  v8f zero = {};
  for (int mt = wave; mt < CC / 16; mt += 8) {
    const int c0 = mt * 16;
    v8f acc[6];
#pragma unroll
    for (int j = 0; j < 6; ++j) acc[j] = zero;

    for (int k0 = 0; k0 < HH; k0 += 4) {
      const int ka = k0 + 2 * hi;
      // A[c][p] = v[b,c,p,w]
      const float* vp = vb + (size_t)(c0 + mr) * HWN + ka * WW + w;
      v2f a;
      a.x = vp[0];
      a.y = vp[WW];
#pragma unroll
      for (int j = 0; j < 6; ++j) {
        // B[p][h] = ast[h*96+p]
        const float* ap = &ast[(j * 16 + mr) * HH + ka];
        v2f bf;
        bf.x = ap[0];
        bf.y = ap[1];
        acc[j] = wmma4(a, bf, acc[j]);
      }
    }

#pragma unroll
    for (int i = 0; i < 8; ++i) {
      const int c = c0 + i + 8 * hi;
#pragma unroll
      for (int j = 0; j < 6; ++j) {
        const int h = j * 16 + mr;
        ob[(size_t)c * HWN + h * WW + w] = acc[j][i];
      }
    }
  }
}

// ---------------------------------------------------------------------------
// Kernel 6: outW + final combine:
// out[b,c,h,w] = gamma*(outH + sum_p v[b,c,h,p]*attW[b,h,w,p]) + x[b,c,h,w].
// ---------------------------------------------------------------------------
__global__ __launch_bounds__(256)
void aggW_kernel(const float* __restrict__ v, const float* __restrict__ attW,
                 const float* __restrict__ x, const float* __restrict__ gamma,
                 float* __restrict__ out) {
  __shared__ float ast[WW * WW];  // ast[w*96+p]
  const int h = blockIdx.x, b = blockIdx.y;
  const int tid = threadIdx.x;
  const float* attb = attW + (((size_t)b * HH + h) * WW) * WW;

  for (int i = tid; i < (WW * WW) / 4; i += 256) {  // 9 full-EXEC iterations
    const int e = i * 4;
    async_ld_lds_b128(&ast[e], attb + e);
  }
  wait_async();
  __syncthreads();

  const float g = gamma[0];
  const int lane = tid & 31, wave = tid >> 5;
  const int mr = lane & 15, hi = lane >> 4;
  const float* vb = v + (size_t)b * CC * HWN;
  const float* xb = x + (size_t)b * CC * HWN;
  float* ob = out + (size_t)b * CC * HWN;

  v8f zero = {};
  for (int mt = wave; mt < CC / 16; mt += 8) {
    const int c0 = mt * 16;
    v8f acc[6];
#pragma unroll
    for (int j = 0; j < 6; ++j) acc[j] = zero;

    for (int k0 = 0; k0 < WW; k0 += 4) {
      const int ka = k0 + 2 * hi;
      // A[c][p] = v[b,c,h,p]  (p contiguous)
      const float* vp = vb + (size_t)(c0 + mr) * HWN + h * WW + ka;
      v2f a;
      a.x = vp[0];
      a.y = vp[1];
#pragma unroll
      for (int j = 0; j < 6; ++j) {
        // B[p][w] = ast[w*96+p]
        const float* ap = &ast[(j * 16 + mr) * WW + ka];
        v2f bf;
        bf.x = ap[0];
        bf.y = ap[1];
        acc[j] = wmma4(a, bf, acc[j]);
      }
    }

#pragma unroll
    for (int i = 0; i < 8; ++i) {
      const int c = c0 + i + 8 * hi;
#pragma unroll
      for (int j = 0; j < 6; ++j) {
        const int w = j * 16 + mr;
        const size_t idx = (size_t)c * HWN + h * WW + w;
        ob[idx] = g * (ob[idx] + acc[j][i]) + xb[idx];
      }
    }
  }
}

// ---------------------------------------------------------------------------
extern "C" void kernel_launch(void* const* d_in, const int* in_sizes, int n_in,
                              void* d_out, int out_size, void* d_ws,
                              size_t ws_size, hipStream_t stream) {
  const float* x  = (const float*)d_in[0];
  const float* Wq = (const float*)d_in[1];
  const float* bq = (const float*)d_in[2];
  const float* Wk = (const float*)d_in[3];
  const float* bk = (const float*)d_in[4];
  const float* Wv = (const float*)d_in[5];
  const float* bv = (const float*)d_in[6];
  const float* gm = (const float*)d_in[7];
  float* out = (float*)d_out;

  // Workspace layout (floats): q | k | v | eH | eW  (~234 MB total)
  float* qws = (float*)d_ws;
  float* kws = qws + (size_t)BB * CQ * HWN;
  float* vws = kws + (size_t)BB * CQ * HWN;
  float* eH  = vws + (size_t)BB * CC * HWN;
  float* eW  = eH  + (size_t)BB * HH * WW * HH;

  qkv_proj_kernel<<<dim3(HWN / 64, 640 / 64, BB), 128, 0, stream>>>(
      x, Wq, bq, Wk, bk, Wv, bv, qws, kws, vws);
  energyH_kernel<<<dim3(WW, BB), 192, 0, stream>>>(qws, kws, eH);
  energyW_kernel<<<dim3(HH, BB), 192, 0, stream>>>(qws, kws, eW);
  softmax_kernel<<<dim3((BB * HH * WW) / 8), 256, 0, stream>>>(eH, eW);
  aggH_kernel<<<dim3(WW, BB), 256, 0, stream>>>(vws, eH, out);
  aggW_kernel<<<dim3(HH, BB), 256, 0, stream>>>(vws, eW, x, gm, out);
}